// DemandExtraction_56298431316372
// MI455X (gfx1250) — compile-verified
//
#include <hip/hip_runtime.h>
#include <math.h>

typedef __attribute__((ext_vector_type(2))) float v2f;
typedef __attribute__((ext_vector_type(8))) float v8f;

#define N_CAT   1000
#define EMB     128
#define ND      8
#define HID     128
#define BB      16
#define LL      50
#define NITEMS  8192

// ---------------- gather: out[r,:] = table[idx[r],:] ----------------
__global__ void gather_rows_kernel(const float* __restrict__ table,
                                   const int* __restrict__ idx,
                                   float* __restrict__ out, int nrows) {
    int e = blockIdx.x * blockDim.x + threadIdx.x;
    if (e >= nrows * EMB) return;
    int r = e >> 7, c = e & 127;
    out[e] = table[idx[r] * EMB + c];
}

// ---------------- generic f32 WMMA GEMM: C = A(Mx128) * B(128xN) (+bias) ----------------
// one wave per 16x16 output tile, 4 waves per block (blockDim.x = 128)
// A,B row-major; K fixed at 128; M multiple of 16; N is a COMPILE-TIME template
// parameter so every B load offset (k*N elements, max 512KB) becomes a 24-bit
// immediate — the unrolled K-loop contains no address arithmetic at all, just
// global_load_b64 (A) / 2x global_load_b32 (B) with immediate offsets + WMMA.
// Branch-free K loop: out-of-range columns are CLAMPED (not zeroed) — column n of
// the WMMA result depends only on column n of B, so clamped lanes only pollute
// accumulator entries that are never stored. EXEC stays all-1s through every WMMA.
template<int N>
__global__ void wmma_gemm_f32_kernel(const float* __restrict__ A,
                                     const float* __restrict__ Bm,
                                     const float* __restrict__ bias,
                                     float* __restrict__ C,
                                     int M) {
    const int lane  = threadIdx.x & 31;
    const int wave  = threadIdx.x >> 5;
    const int ntile = blockIdx.x * 4 + wave;
    const int mtile = blockIdx.y;
    if (ntile * 16 >= N || mtile * 16 >= M) return;   // wave-uniform exit

    const int half = lane >> 4;        // 0: lanes 0-15, 1: lanes 16-31
    const int r    = lane & 15;
    const int n    = ntile * 16 + r;
    const int ncl  = (n < N) ? n : (N - 1);           // clamped, branch-free loads

    // A 16x4 tile per step: lane holds A[r][k+2*half], A[r][k+2*half+1]
    const float2* A2 = (const float2*)(A + (size_t)(mtile * 16 + r) * 128) + half;
    // B 4x16 tile per step: lane holds B[k+2*half][n], B[k+2*half+1][n]
    const float* Bp = Bm + (size_t)(2 * half) * N + ncl;   // lane base; rest is immediates

    v8f acc = {};
#pragma unroll
    for (int k = 0; k < 128; k += 4) {
        float2 av = A2[k >> 1];                        // immediate offset (k*4 B)
        v2f a; a.x = av.x; a.y = av.y;
        v2f b;
        b.x = Bp[(size_t)k * N];                       // immediate offset (k*N*4 B)
        b.y = Bp[(size_t)k * N + N];                   // immediate offset ((k+1)*N*4 B)
        acc = __builtin_amdgcn_wmma_f32_16x16x4_f32(false, a, false, b,
                                                    (short)0, acc, false, false);
    }

    if (n < N) {
        float bv = bias ? bias[n] : 0.0f;
#pragma unroll
        for (int v = 0; v < 8; ++v) {
            int row = mtile * 16 + half * 8 + v;      // C/D layout: M = v + 8*half
            if (row < M) C[(size_t)row * N + n] = acc[v] + bv;
        }
    }
}

// ---------------- logsumexp over L: agg[b,d,h] = LSE_l hd[b,l,d*128+h] ----------------
__global__ void lse_kernel(const float* __restrict__ hd, float* __restrict__ agg) {
    int idx = blockIdx.x * blockDim.x + threadIdx.x;   // 16*1024
    if (idx >= BB * ND * HID) return;
    int b   = idx >> 10;
    int rem = idx & 1023;                               // d*128 + h
    const float* p = hd + (size_t)b * LL * (ND * HID) + rem;
    float m = -__builtin_inff();
    for (int l = 0; l < LL; ++l) m = fmaxf(m, p[(size_t)l * (ND * HID)]);
    float s = 0.0f;
    for (int l = 0; l < LL; ++l) s += expf(p[(size_t)l * (ND * HID)] - m);
    agg[idx] = logf(s) + m;
}

// ---------------- per-batch off-diagonal cosine-sim partial sums ----------------
__global__ void sim_kernel(const float* __restrict__ agg, float* __restrict__ part) {
    __shared__ float sA[ND * HID];
    __shared__ float invn[ND];
    __shared__ float pairs[28];
    int b = blockIdx.x;
    const float* ab = agg + (size_t)b * ND * HID;
    for (int i = threadIdx.x; i < ND * HID; i += blockDim.x) sA[i] = ab[i];
    __syncthreads();
    if (threadIdx.x < ND) {
        float s = 0.0f;
        for (int hh = 0; hh < HID; ++hh) { float v = sA[threadIdx.x * HID + hh]; s += v * v; }
        invn[threadIdx.x] = 1.0f / (sqrtf(s) + 1e-12f);
    }
    __syncthreads();
    if (threadIdx.x < 28) {
        int t = threadIdx.x, d = 0, e = 0, c = t;
        for (d = 0; d < ND; ++d) { int cnt = ND - 1 - d; if (c < cnt) { e = d + 1 + c; break; } c -= cnt; }
        float s = 0.0f;
        for (int hh = 0; hh < HID; ++hh) s += sA[d * HID + hh] * sA[e * HID + hh];
        pairs[t] = s * invn[d] * invn[e];
    }
    __syncthreads();
    if (threadIdx.x == 0) {
        float s = 0.0f;
        for (int t = 0; t < 28; ++t) s += pairs[t];
        part[b] = 2.0f * s;                     // off-diag = 2 * sum_{d<e}
    }
}

__global__ void loss_kernel(const float* __restrict__ part, float* __restrict__ out) {
    if (threadIdx.x == 0 && blockIdx.x == 0) {
        float s = 0.0f;
        for (int b = 0; b < BB; ++b) s += part[b];
        out[0] = s / (float)(BB * ND * (ND - 1));
    }
}

// ---------------- demand_score: relu(pd + pk + b1) . w_score ----------------
// one block (128 threads) per (b,l,d); output flat (B,L,D)
__global__ void dscore_kernel(const float* __restrict__ pd, const float* __restrict__ pk,
                              const float* __restrict__ b1, const float* __restrict__ w,
                              float* __restrict__ out) {
    __shared__ float red[HID];
    int z = blockIdx.x;                 // b*L*D + l*D + d
    int b = z / (LL * ND);
    int l = (z / ND) % LL;
    int d = z % ND;
    int hh = threadIdx.x;
    float v = fmaxf(pd[(size_t)(b * ND + d) * HID + hh] +
                    pk[(size_t)(b * LL + l) * HID + hh] + b1[hh], 0.0f) * w[hh];
    red[hh] = v;
    __syncthreads();
    for (int s = HID / 2; s > 0; s >>= 1) {
        if (hh < s) red[hh] += red[hh + s];
        __syncthreads();
    }
    if (hh == 0) out[z] = red[0];
}

// ---------------- demand_score_candidate: relu(pd + pkc + b1) . w_score ----------------
// grid (NITEMS/256, B*D), block 256
__global__ void cscore_kernel(const float* __restrict__ pd, const float* __restrict__ pkc,
                              const float* __restrict__ b1, const float* __restrict__ w,
                              float* __restrict__ out) {
    __shared__ float pdb[HID];
    __shared__ float ws[HID];
    int bd  = blockIdx.y;               // b*D + d
    int tid = threadIdx.x;
    if (tid < HID) { pdb[tid] = pd[(size_t)bd * HID + tid] + b1[tid]; ws[tid] = w[tid]; }
    __syncthreads();
    int i = blockIdx.x * blockDim.x + tid;
    const float4* row = (const float4*)(pkc + (size_t)i * HID);
    float acc = 0.0f;
#pragma unroll 4
    for (int q = 0; q < HID / 4; ++q) {
        float4 v = row[q];
        int hh = q * 4;
        acc += fmaxf(pdb[hh + 0] + v.x, 0.0f) * ws[hh + 0];
        acc += fmaxf(pdb[hh + 1] + v.y, 0.0f) * ws[hh + 1];
        acc += fmaxf(pdb[hh + 2] + v.z, 0.0f) * ws[hh + 2];
        acc += fmaxf(pdb[hh + 3] + v.w, 0.0f) * ws[hh + 3];
    }
    out[(size_t)bd * NITEMS + i] = acc;
}

extern "C" void kernel_launch(void* const* d_in, const int* in_sizes, int n_in,
                              void* d_out, int out_size, void* d_ws, size_t ws_size,
                              hipStream_t stream) {
    const int*   inp   = (const int*)d_in[0];     // (B,L)
    const int*   cand  = (const int*)d_in[1];     // (NITEMS,)
    // d_in[2] session_last_catgy_index, d_in[3] mask_catgy: unused by reference
    const float* emb_t = (const float*)d_in[4];   // (1000,128)
    const float* Wd    = (const float*)d_in[5];   // (128,1024)
    const float* Wk    = (const float*)d_in[6];   // (128,128)
    const float* bk    = (const float*)d_in[7];   // (128,)
    const float* W1    = (const float*)d_in[8];   // (256,128): Wd1 = W1, Wk1 = W1+128*128
    const float* b1    = (const float*)d_in[9];   // (128,)
    const float* wsc   = (const float*)d_in[10];  // (128,)
    const float* Wc    = (const float*)d_in[11];  // (128,1000)
    const float* bc    = (const float*)d_in[12];  // (1000,)
    float* out = (float*)d_out;

    // output regions (flat, return order)
    float* out_catgy = out;                                   // 16*8*1000
    float* out_ds    = out + 128000;                          // 16*50*8 (view-reshaped)
    float* out_dsc   = out + 134400;                          // 16*8*8192
    float* out_emb   = out + 1182976;                         // 16*50*128
    float* out_embc  = out + 1285376;                         // 8192*128
    float* out_loss  = out + 2333952;                         // scalar

    // workspace regions (floats)
    float* ws   = (float*)d_ws;
    float* hd   = ws;                   // 800*1024
    float* agg  = ws + 819200;          // 128*128
    float* hk   = ws + 835584;          // 800*128
    float* hkc  = ws + 937984;          // 8192*128
    float* pd   = ws + 1986560;         // 128*128
    float* pk   = ws + 2002944;         // 800*128
    float* pkc  = ws + 2105344;         // 8192*128
    float* part = ws + 3153920;         // 16

    const float* Wk1 = W1 + 128 * 128;

    // 1) embedding gathers (also outputs)
    gather_rows_kernel<<<(BB * LL * EMB + 255) / 256, 256, 0, stream>>>(emb_t, inp, out_emb, BB * LL);
    gather_rows_kernel<<<(NITEMS * EMB + 255) / 256, 256, 0, stream>>>(emb_t, cand, out_embc, NITEMS);

    // 2) hidden_demand = emb @ Wd  (800 x 1024)
    wmma_gemm_f32_kernel<1024><<<dim3(16, 50), 128, 0, stream>>>(out_emb, Wd, nullptr, hd, 800);

    // 3) agg = logsumexp_L(hidden_demand)
    lse_kernel<<<(BB * ND * HID + 255) / 256, 256, 0, stream>>>(hd, agg);

    // 4) hidden_key = emb @ Wk + bk ; hidden_key_cand = emb_cand @ Wk + bk
    wmma_gemm_f32_kernel<128><<<dim3(2, 50),  128, 0, stream>>>(out_emb,  Wk, bk, hk,  800);
    wmma_gemm_f32_kernel<128><<<dim3(2, 512), 128, 0, stream>>>(out_embc, Wk, bk, hkc, 8192);

    // 5) pd = agg @ Wd1 ; pk = hk @ Wk1 ; pkc = hkc @ Wk1
    wmma_gemm_f32_kernel<128><<<dim3(2, 8),   128, 0, stream>>>(agg, W1,  nullptr, pd,  128);
    wmma_gemm_f32_kernel<128><<<dim3(2, 50),  128, 0, stream>>>(hk,  Wk1, nullptr, pk,  800);
    wmma_gemm_f32_kernel<128><<<dim3(2, 512), 128, 0, stream>>>(hkc, Wk1, nullptr, pkc, 8192);

    // 6) catgy_score = agg @ Wc + bc  (128 x 1000, N-edge clamped)
    wmma_gemm_f32_kernel<1000><<<dim3(16, 8), 128, 0, stream>>>(agg, Wc, bc, out_catgy, 128);

    // 7) demand_sim_loss
    sim_kernel<<<BB, 128, 0, stream>>>(agg, part);
    loss_kernel<<<1, 32, 0, stream>>>(part, out_loss);

    // 8) demand_score
    dscore_kernel<<<BB * LL * ND, HID, 0, stream>>>(pd, pk, b1, wsc, out_ds);

    // 9) demand_score_candidate
    cscore_kernel<<<dim3(NITEMS / 256, BB * ND), 256, 0, stream>>>(pd, pkc, b1, wsc, out_dsc);
}